// fab_penalty_ls_curve_72602127171972
// MI455X (gfx1250) — compile-verified
//
#include <hip/hip_runtime.h>
#include <math.h>

// ---------------------------------------------------------------------------
// Separable reformulation of the level-set curvature penalty.
//   K = Ky (x) Kx  (Kronecker, tensor grid + separable Gaussian)
//   model = K^-1 p        <=>  M   = Ky^-1 P Kx^-1      (50x50 each)
//   phi   = G^T model     <=>  Phi = Gy^T M Gx          (Gy,Gx: 50x200)
// All GEMMs run on V_WMMA_F32_16X16X4_F32 with zero-padded tiles
// (50 -> 64, 200 -> 208). Everything fp32 (solve is ill-conditioned).
// ---------------------------------------------------------------------------

typedef __attribute__((ext_vector_type(2))) float v2f;
typedef __attribute__((ext_vector_type(8))) float v8f;

#define PI_F 3.14159265358979323846f

// -------------------- small Gaussian kernel matrices -----------------------
__global__ void gauss_pad_kernel(const float* __restrict__ a, int na,
                                 const float* __restrict__ b, int nb,
                                 int PA, int PB,
                                 const float* __restrict__ sigma_p,
                                 float* __restrict__ out) {
    int idx = blockIdx.x * blockDim.x + threadIdx.x;
    if (idx >= PA * PB) return;
    int i = idx / PB, j = idx % PB;
    float v = 0.0f;
    if (i < na && j < nb) {
        float s = sigma_p[0];
        float dd = a[i] - b[j];
        v = expf(-(dd * dd) / (2.0f * s * s));
    }
    out[idx] = v;
}

__global__ void pack_params_kernel(const float* __restrict__ p, int n, int P,
                                   float* __restrict__ out) {
    int idx = blockIdx.x * blockDim.x + threadIdx.x;
    if (idx >= P * P) return;
    int i = idx / P, j = idx % P;
    out[idx] = (i < n && j < n) ? p[i * n + j] : 0.0f;
}

// -------------------- 50x50 Gauss-Jordan inverse in LDS --------------------
#define MAXR 64
__global__ __launch_bounds__(256) void invert_gauss_kernel(
    const float* __restrict__ y_rho, const float* __restrict__ x_rho,
    int n, int P, const float* __restrict__ sigma_p,
    float* __restrict__ Ainv, float* __restrict__ Binv) {
    __shared__ float Aug[MAXR * 2 * MAXR];  // stride 128, 32 KB
    __shared__ float colk[MAXR];
    __shared__ float pivs;
    const float* c  = (blockIdx.x == 0) ? y_rho : x_rho;
    float* outp     = (blockIdx.x == 0) ? Ainv  : Binv;
    const float s   = sigma_p[0];
    const float i2s = 1.0f / (2.0f * s * s);
    const int W = 2 * n, stride = 2 * MAXR;
    const int tid = threadIdx.x, nt = blockDim.x;

    for (int idx = tid; idx < n * W; idx += nt) {
        int i = idx / W, j = idx % W;
        float v;
        if (j < n) { float dd = c[i] - c[j]; v = expf(-dd * dd * i2s); }
        else        v = (j - n == i) ? 1.0f : 0.0f;
        Aug[i * stride + j] = v;
    }
    __syncthreads();
    for (int k = 0; k < n; ++k) {             // SPD -> no pivoting
        if (tid == 0) pivs = 1.0f / Aug[k * stride + k];
        __syncthreads();
        float pv = pivs;
        for (int j = tid; j < W; j += nt) Aug[k * stride + j] *= pv;
        __syncthreads();
        for (int i = tid; i < n; i += nt) colk[i] = Aug[i * stride + k];
        __syncthreads();
        for (int idx = tid; idx < n * W; idx += nt) {
            int i = idx / W, j = idx % W;
            if (i != k) Aug[i * stride + j] -= colk[i] * Aug[k * stride + j];
        }
        __syncthreads();
    }
    for (int idx = tid; idx < P * P; idx += nt) {
        int i = idx / P, j = idx % P;
        outp[idx] = (i < n && j < n) ? Aug[i * stride + (n + j)] : 0.0f;
    }
}

// -------------------- fp32 WMMA GEMM: C = A*B (or A^T*B) -------------------
// One 16x16 C tile per wave; K loop in steps of 4 over V_WMMA_F32_16X16X4_F32,
// unrolled 4x so 8 fragment loads issue ahead of 4 chained WMMAs.
// Fragment layouts (wave32): A 16x4: M=l%16, K=r+2*(l/16); B 4x16:
// K=r+2*(l/16), N=l%16; C/D: row=r+8*(l/16), col=l%16.
template <bool TRANSA>
__global__ __launch_bounds__(128) void wmma_gemm_f32_kernel(
    const float* __restrict__ A, const float* __restrict__ B,
    float* __restrict__ C, int Mdim, int Kdim, int Ndim) {
    const int lane = threadIdx.x & 31;
    const int wave = threadIdx.x >> 5;
    const int tilesN = Ndim >> 4;
    const int ntiles = (Mdim >> 4) * tilesN;
    const int tile = blockIdx.x * 4 + wave;   // wave-uniform
    if (tile >= ntiles) return;
    const int tm = tile / tilesN, tn = tile % tilesN;
    const int lr = lane & 15, half = lane >> 4;
    const int row = tm * 16 + lr, col = tn * 16 + lr;

    // Hoisted base pointers; 32-bit offsets in the loop.
    const float* __restrict__ Arow = TRANSA ? (A + row) : (A + row * Kdim);
    const float* __restrict__ Bcol = B + col;
    const int h2 = 2 * half;

    v8f acc = {0.f, 0.f, 0.f, 0.f, 0.f, 0.f, 0.f, 0.f};

    auto loadA = [&](int k) -> v2f {         // k = k0 + 2*half
        if (TRANSA) {
            v2f a;
            a.x = Arow[k * Mdim];
            a.y = Arow[(k + 1) * Mdim];
            return a;
        } else {
            return *reinterpret_cast<const v2f*>(Arow + k);  // 8B aligned
        }
    };
    auto loadB = [&](int k) -> v2f {
        v2f b;
        b.x = Bcol[k * Ndim];
        b.y = Bcol[(k + 1) * Ndim];
        return b;
    };

    int k0 = 0;
    for (; k0 + 16 <= Kdim; k0 += 16) {
        v2f a0 = loadA(k0 + h2);
        v2f b0 = loadB(k0 + h2);
        v2f a1 = loadA(k0 + 4 + h2);
        v2f b1 = loadB(k0 + 4 + h2);
        v2f a2 = loadA(k0 + 8 + h2);
        v2f b2 = loadB(k0 + 8 + h2);
        v2f a3 = loadA(k0 + 12 + h2);
        v2f b3 = loadB(k0 + 12 + h2);
        acc = __builtin_amdgcn_wmma_f32_16x16x4_f32(false, a0, false, b0, (short)0, acc, false, false);
        acc = __builtin_amdgcn_wmma_f32_16x16x4_f32(false, a1, false, b1, (short)0, acc, false, false);
        acc = __builtin_amdgcn_wmma_f32_16x16x4_f32(false, a2, false, b2, (short)0, acc, false, false);
        acc = __builtin_amdgcn_wmma_f32_16x16x4_f32(false, a3, false, b3, (short)0, acc, false, false);
    }
    for (; k0 < Kdim; k0 += 4) {
        v2f a = loadA(k0 + h2);
        v2f b = loadB(k0 + h2);
        acc = __builtin_amdgcn_wmma_f32_16x16x4_f32(false, a, false, b, (short)0, acc, false, false);
    }

    float* __restrict__ Crow = C + (tm * 16 + half * 8) * Ndim + col;
#pragma unroll
    for (int r = 0; r < 8; ++r)
        Crow[r * Ndim] = acc[r];
}

// -------------------- eps = 0.5*(tanh(0.1*phi)+1), mirrored ----------------
__global__ void eps_kernel(const float* __restrict__ Phi, int PP, int np,
                           float* __restrict__ eps) {
    int NY = 2 * np;
    int idx = blockIdx.x * blockDim.x + threadIdx.x;
    if (idx >= np * NY) return;
    int i = idx / NY, j = idx % NY;
    int jj = (j < np) ? j : (NY - 1 - j);
    float ph = Phi[(size_t)i * PP + jj];
    eps[idx] = 0.5f * (tanhf(0.1f * ph) + 1.0f);
}

// -------------------- stencil helpers (match _grad exactly) ----------------
__device__ __forceinline__ float grad0(const float* __restrict__ e, int i, int j,
                                       int NX, int NY, float d) {
    if (i == 0)      return (e[NY + j] - e[j]) / d;
    if (i == NX - 1) return (e[(size_t)(NX - 1) * NY + j] - e[(size_t)(NX - 2) * NY + j]) / d;
    return (e[(size_t)(i + 1) * NY + j] - e[(size_t)(i - 1) * NY + j]) / (2.0f * d);
}
__device__ __forceinline__ float grad1(const float* __restrict__ e, int i, int j,
                                       int NY, float d) {
    const float* r = e + (size_t)i * NY;
    if (j == 0)      return (r[1] - r[0]) / d;
    if (j == NY - 1) return (r[NY - 1] - r[NY - 2]) / d;
    return (r[j + 1] - r[j - 1]) / (2.0f * d);
}

__global__ void d1_kernel(const float* __restrict__ eps, int NX, int NY,
                          const float* __restrict__ d_p,
                          float* __restrict__ ex, float* __restrict__ ey) {
    int idx = blockIdx.x * blockDim.x + threadIdx.x;
    if (idx >= NX * NY) return;
    int i = idx / NY, j = idx % NY;
    float d = d_p[0];
    ex[idx] = grad0(eps, i, j, NX, NY, d) + 1e-12f;
    ey[idx] = grad1(eps, i, j, NY, d) + 1e-12f;
}

// -------------------- curvature + deterministic block reduce ---------------
__global__ __launch_bounds__(256) void curv_kernel(
    const float* __restrict__ eps, const float* __restrict__ ex,
    const float* __restrict__ ey, int NX, int NY,
    const float* __restrict__ d_p, float* __restrict__ partials) {
    __shared__ float red[256];
    int idx = blockIdx.x * 256 + threadIdx.x;
    float v = 0.0f;
    if (idx < NX * NY) {
        int i = idx / NY, j = idx % NY;
        float d = d_p[0];
        float exx = grad0(ex, i, j, NX, NY, d);
        float exy = grad1(ex, i, j, NY, d);
        float eyy = grad1(ey, i, j, NY, d);
        float gx = ex[idx], gy = ey[idx];
        float epsv = fmaxf(sqrtf(gx * gx + gy * gy), 1e-32f / 6.0f);
        float kc = (gx * gx * eyy - 2.0f * gx * gy * exy + gy * gy * exx) /
                   (epsv * epsv * epsv);
        float cc = fabsf(kc * atanf(epsv / eps[idx])) - (PI_F / 1.1f);
        v = fmaxf(cc, 0.0f);           // fmaxf(NaN,0)=0  ==  nansum drop
        if (isnan(v)) v = 0.0f;
    }
    red[threadIdx.x] = v;
    __syncthreads();
    for (int s = 128; s > 0; s >>= 1) {
        if (threadIdx.x < s) red[threadIdx.x] += red[threadIdx.x + s];
        __syncthreads();
    }
    if (threadIdx.x == 0) partials[blockIdx.x] = red[0];
}

__global__ void final_kernel(const float* __restrict__ partials, int n,
                             const float* __restrict__ gs_p,
                             float* __restrict__ out) {
    if (threadIdx.x == 0 && blockIdx.x == 0) {
        float s = 0.0f;
        for (int i = 0; i < n; ++i) s += partials[i];   // fixed order
        float gs = gs_p[0];
        out[0] = 1.0f /*ALPHA*/ * s * gs * gs;
    }
}

// ---------------------------------------------------------------------------
extern "C" void kernel_launch(void* const* d_in, const int* in_sizes, int n_in,
                              void* d_out, int out_size, void* d_ws, size_t ws_size,
                              hipStream_t stream) {
    const float* params    = (const float*)d_in[0];
    const float* x_rho     = (const float*)d_in[1];
    const float* y_rho     = (const float*)d_in[2];
    const float* x_phi     = (const float*)d_in[3];
    const float* y_phi     = (const float*)d_in[4];
    const float* rho_size  = (const float*)d_in[9];
    const float* grid_size = (const float*)d_in[10];
    float* out = (float*)d_out;

    const int nr = in_sizes[1];             // 50
    const int np = in_sizes[3];             // 200
    const int PR = (nr + 15) & ~15;         // 64
    const int PP = (np + 15) & ~15;         // 208
    const int NX = np, NY = 2 * np;         // 200 x 400

    float* ws = (float*)d_ws;
    size_t o = 0;
    float* Gy   = ws + o; o += (size_t)PR * PP;
    float* Gx   = ws + o; o += (size_t)PR * PP;
    float* Pm   = ws + o; o += (size_t)PR * PR;
    float* Ainv = ws + o; o += (size_t)PR * PR;
    float* Binv = ws + o; o += (size_t)PR * PR;
    float* M1   = ws + o; o += (size_t)PR * PR;
    float* Mm   = ws + o; o += (size_t)PR * PR;
    float* T    = ws + o; o += (size_t)PR * PP;
    float* Phi  = ws + o; o += (size_t)PP * PP;
    float* eps  = ws + o; o += (size_t)NX * NY;
    float* ex   = ws + o; o += (size_t)NX * NY;
    float* ey   = ws + o; o += (size_t)NX * NY;
    const int nelem  = NX * NY;
    const int nparts = (nelem + 255) / 256;
    float* partials = ws + o; o += (size_t)nparts;
    (void)ws_size; (void)n_in; (void)out_size;  // ~1.4 MB total, fits d_ws

    const int B = 256;
    // kernel matrices Gy, Gx and packed params
    gauss_pad_kernel<<<(PR * PP + B - 1) / B, B, 0, stream>>>(y_rho, nr, y_phi, np, PR, PP, rho_size, Gy);
    gauss_pad_kernel<<<(PR * PP + B - 1) / B, B, 0, stream>>>(x_rho, nr, x_phi, np, PR, PP, rho_size, Gx);
    pack_params_kernel<<<(PR * PR + B - 1) / B, B, 0, stream>>>(params, nr, PR, Pm);
    // Ky^-1 and Kx^-1 (one workgroup each)
    invert_gauss_kernel<<<2, 256, 0, stream>>>(y_rho, x_rho, nr, PR, rho_size, Ainv, Binv);

    // GEMM chain on fp32 WMMA: M = Ky^-1 P Kx^-1 ; T = M Gx ; Phi = Gy^T T
    auto blocks = [](int m, int n) { return ((m >> 4) * (n >> 4) + 3) / 4; };
    wmma_gemm_f32_kernel<false><<<blocks(PR, PR), 128, 0, stream>>>(Ainv, Pm,  M1, PR, PR, PR);
    wmma_gemm_f32_kernel<false><<<blocks(PR, PR), 128, 0, stream>>>(M1,  Binv, Mm, PR, PR, PR);
    wmma_gemm_f32_kernel<false><<<blocks(PR, PP), 128, 0, stream>>>(Mm,  Gx,   T,  PR, PR, PP);
    wmma_gemm_f32_kernel<true ><<<blocks(PP, PP), 128, 0, stream>>>(Gy,  T,   Phi, PP, PR, PP);

    // eps -> gradients -> curvature penalty -> deterministic nansum
    eps_kernel<<<(nelem + B - 1) / B, B, 0, stream>>>(Phi, PP, np, eps);
    d1_kernel<<<(nelem + B - 1) / B, B, 0, stream>>>(eps, NX, NY, grid_size, ex, ey);
    curv_kernel<<<nparts, 256, 0, stream>>>(eps, ex, ey, NX, NY, grid_size, partials);
    final_kernel<<<1, 64, 0, stream>>>(partials, nparts, grid_size, out);
}